// LocalAveragePoolingSegmenter_11922829214088
// MI455X (gfx1250) — compile-verified
//
#include <hip/hip_runtime.h>
#include <hip/hip_bf16.h>
#include <stdint.h>

// Problem constants (from reference)
#define B_   16
#define TF   4096
#define D_   512
#define TT   1024
#define NCHUNK 32
#define CHUNK  128   // TF / NCHUNK
#define DV   (D_ / 4)   // 128 float4s per row

static constexpr size_t P_ELEMS = (size_t)B_ * (TF + 1) * D_;   // exclusive prefix sums
static constexpr size_t S_ELEMS = (size_t)B_ * NCHUNK * D_;     // chunk sums -> chunk prefix

#define AS1 __attribute__((address_space(1)))
#define AS3 __attribute__((address_space(3)))

typedef int v4i __attribute__((ext_vector_type(4)));

#if defined(__gfx1250__) && __has_builtin(__builtin_amdgcn_global_load_async_to_lds_b128)
#define HAVE_ASYNC_LDS 1
#else
#define HAVE_ASYNC_LDS 0
#endif

__device__ __forceinline__ void wait_asynccnt0() {
#if __has_builtin(__builtin_amdgcn_s_wait_asynccnt)
    __builtin_amdgcn_s_wait_asynccnt(0);
#else
    asm volatile("s_wait_asynccnt 0" ::: "memory");
#endif
}

__device__ __forceinline__ void acc4(float4& a, const float4 b) {
    a.x += b.x; a.y += b.y; a.z += b.z; a.w += b.w;
}

// ---------------------------------------------------------------------------
// Kernel 1: chunk sums. grid = B*NCHUNK blocks, 128 threads (float4 per lane).
// Streams feats once (128 MB) with b128 loads (512 B / wave / instruction).
// ---------------------------------------------------------------------------
__global__ void k_chunk_sums(const float* __restrict__ feats, float* __restrict__ S) {
    const int blk = blockIdx.x;
    const int b = blk / NCHUNK, c = blk % NCHUNK;
    const int q = threadIdx.x;                   // 0..127 (float4 index in row)
    const float4* p = (const float4*)(feats + ((size_t)b * TF + (size_t)c * CHUNK) * D_) + q;
    float4 acc = make_float4(0.f, 0.f, 0.f, 0.f);
#pragma unroll 8
    for (int f = 0; f < CHUNK; ++f) {
        if ((f & 7) == 0 && f + 16 < CHUNK)
            __builtin_prefetch(p + (size_t)(f + 16) * DV, 0, 1);  // global_prefetch_b8
        acc4(acc, p[(size_t)f * DV]);
    }
    ((float4*)(S + ((size_t)b * NCHUNK + c) * D_))[q] = acc;
}

// ---------------------------------------------------------------------------
// Kernel 2: exclusive prefix over the 32 chunk sums, in place.
// One thread per (b,d) column; columns are disjoint -> no races.
// ---------------------------------------------------------------------------
__global__ void k_chunk_prefix(float* __restrict__ S) {
    const int i = blockIdx.x * blockDim.x + threadIdx.x;
    if (i >= B_ * D_) return;
    const int b = i / D_, d = i % D_;
    float run = 0.f;
#pragma unroll
    for (int c = 0; c < NCHUNK; ++c) {
        const size_t idx = ((size_t)b * NCHUNK + c) * D_ + d;
        const float v = S[idx];
        S[idx] = run;
        run += v;
    }
}

// ---------------------------------------------------------------------------
// Kernel 3: write full exclusive prefix P[b][f][d], f in [0, TF], seeded by
// the chunk prefix. grid = B*NCHUNK blocks, 128 threads (float4 per lane).
// b128 loads + b128 stores.
// ---------------------------------------------------------------------------
__global__ void k_prefix(const float* __restrict__ feats,
                         const float* __restrict__ S,
                         float* __restrict__ P) {
    const int blk = blockIdx.x;
    const int b = blk / NCHUNK, c = blk % NCHUNK;
    const int q = threadIdx.x;                   // 0..127
    float4 run = ((const float4*)(S + ((size_t)b * NCHUNK + c) * D_))[q];
    const float4* fp = (const float4*)(feats + ((size_t)b * TF + (size_t)c * CHUNK) * D_) + q;
    float4* pp = (float4*)(P + ((size_t)b * (TF + 1) + (size_t)c * CHUNK) * D_) + q;
#pragma unroll 4
    for (int j = 0; j < CHUNK; ++j) {
        if ((j & 7) == 0 && j + 16 < CHUNK)
            __builtin_prefetch(fp + (size_t)(j + 16) * DV, 0, 1);
        pp[(size_t)j * DV] = run;
        acc4(run, fp[(size_t)j * DV]);
    }
    if (c == NCHUNK - 1) pp[(size_t)CHUNK * DV] = run;   // P[b][TF][:] = total
}

// ---------------------------------------------------------------------------
// Kernel 4: per-token gather: out = (P[e1] - P[s]) / (e1 - s), or 0 if the
// token is padding. grid = B*TT blocks, 128 threads, float4 per lane.
// Rows of P are staged through LDS with CDNA5 async global->LDS copies.
// ---------------------------------------------------------------------------
__global__ void k_gather(const float* __restrict__ P,
                         const int*   __restrict__ feat_len,
                         const int*   __restrict__ tok_len,
                         const float* __restrict__ align,
                         float*       __restrict__ out) {
    const int blk = blockIdx.x;            // b*TT + t
    const int b = blk / TT, t = blk % TT;
    const int i = threadIdx.x;             // 0..127 (float4 index)

    const int   fl = feat_len[b];
    const float a0 = align[(size_t)blk * 2 + 0];
    const float a1 = align[(size_t)blk * 2 + 1];
    int s = (int)(a0 * (float)fl);         // trunc toward zero, matches .astype(int32)
    int e = (int)(a1 * (float)fl);
    s = min(max(s, 0), TF);
    e = min(max(e, 0), TF);
    const int e1 = min(e + 1, TF);
    const bool valid = (t < tok_len[b]) && (e1 > s);
    const float inv = valid ? 1.0f / (float)(e1 - s) : 0.0f;  // invalid/empty -> 0 (nan_to_num)

    const float4* hi = (const float4*)(P + ((size_t)b * (TF + 1) + (size_t)e1) * D_);
    const float4* lo = (const float4*)(P + ((size_t)b * (TF + 1) + (size_t)s ) * D_);
    float4* o = (float4*)(out + (size_t)blk * D_);

    float4 h, l;
#if HAVE_ASYNC_LDS
    __shared__ float4 sHi[128];
    __shared__ float4 sLo[128];
    __builtin_amdgcn_global_load_async_to_lds_b128(
        (AS1 v4i*)(hi + i), (AS3 v4i*)(sHi + i), 0, 0);
    __builtin_amdgcn_global_load_async_to_lds_b128(
        (AS1 v4i*)(lo + i), (AS3 v4i*)(sLo + i), 0, 0);
    wait_asynccnt0();
    __syncthreads();
    h = sHi[i];
    l = sLo[i];
#else
    h = hi[i];
    l = lo[i];
#endif
    float4 r;
    r.x = (h.x - l.x) * inv;
    r.y = (h.y - l.y) * inv;
    r.z = (h.z - l.z) * inv;
    r.w = (h.w - l.w) * inv;
    o[i] = r;
}

// ---------------------------------------------------------------------------
// Fallback (workspace too small): direct segment summation per token.
// grid = B*TT blocks, 128 threads (float4 per lane). Slow but correct.
// ---------------------------------------------------------------------------
__global__ void k_direct(const float* __restrict__ feats,
                         const int*   __restrict__ feat_len,
                         const int*   __restrict__ tok_len,
                         const float* __restrict__ align,
                         float*       __restrict__ out) {
    const int blk = blockIdx.x;
    const int b = blk / TT, t = blk % TT;
    const int q = threadIdx.x;             // 0..127

    const int   fl = feat_len[b];
    const float a0 = align[(size_t)blk * 2 + 0];
    const float a1 = align[(size_t)blk * 2 + 1];
    int s = (int)(a0 * (float)fl);
    int e = (int)(a1 * (float)fl);
    s = min(max(s, 0), TF);
    e = min(max(e, 0), TF);
    const int e1 = min(e + 1, TF);
    const bool valid = (t < tok_len[b]) && (e1 > s);
    const float inv = valid ? 1.0f / (float)(e1 - s) : 0.0f;

    float4 acc = make_float4(0.f, 0.f, 0.f, 0.f);
    const float4* fp = (const float4*)(feats + (size_t)b * TF * D_) + q;
    for (int f = s; f < e1; ++f)
        acc4(acc, fp[(size_t)f * DV]);
    float4 r;
    r.x = acc.x * inv; r.y = acc.y * inv; r.z = acc.z * inv; r.w = acc.w * inv;
    ((float4*)(out + (size_t)blk * D_))[q] = r;
}

// ---------------------------------------------------------------------------
// Tail: second tuple output (asr_token_lengths) appended as float values.
// ---------------------------------------------------------------------------
__global__ void k_tail(const int* __restrict__ tok_len, float* __restrict__ out) {
    const int i = threadIdx.x;
    if (i < B_) out[(size_t)B_ * TT * D_ + i] = (float)tok_len[i];
}

extern "C" void kernel_launch(void* const* d_in, const int* in_sizes, int n_in,
                              void* d_out, int out_size, void* d_ws, size_t ws_size,
                              hipStream_t stream) {
    const float* feats = (const float*)d_in[0];      // [B, TF, D] f32
    const int*   flen  = (const int*)  d_in[1];      // [B] i32
    /* d_in[2] = asr_token_ids (i64) — unused by the math */
    const int*   tlen  = (const int*)  d_in[3];      // [B] i32
    const float* algn  = (const float*)d_in[4];      // [B, TT, 2] f32
    float* out = (float*)d_out;

    const size_t need = (P_ELEMS + S_ELEMS) * sizeof(float);
    if (ws_size >= need) {
        float* P = (float*)d_ws;
        float* S = P + P_ELEMS;
        k_chunk_sums  <<<dim3(B_ * NCHUNK),           dim3(128), 0, stream>>>(feats, S);
        k_chunk_prefix<<<dim3((B_ * D_ + 255) / 256), dim3(256), 0, stream>>>(S);
        k_prefix      <<<dim3(B_ * NCHUNK),           dim3(128), 0, stream>>>(feats, S, P);
        k_gather      <<<dim3(B_ * TT),               dim3(128), 0, stream>>>(P, flen, tlen, algn, out);
    } else {
        k_direct      <<<dim3(B_ * TT),               dim3(128), 0, stream>>>(feats, flen, tlen, algn, out);
    }
    k_tail<<<dim3(1), dim3(32), 0, stream>>>(tlen, out);
}